// MultiResGrid2D_88407606821460
// MI455X (gfx1250) — compile-verified
//
#include <hip/hip_runtime.h>

typedef float v4f __attribute__((ext_vector_type(4)));
typedef float v2f __attribute__((ext_vector_type(2)));

#define C_CH 16
#define NLEV 5
#define TP_PAD 258  // 258 % 64 == 2 -> conflict-free LDS reads in transpose

#if defined(__has_builtin)
#if __has_builtin(__builtin_amdgcn_global_load_async_to_lds_b32) && \
    __has_builtin(__builtin_amdgcn_s_wait_asynccnt)
#define USE_ASYNC_LDS 1
#endif
#endif

#ifdef USE_ASYNC_LDS
typedef __attribute__((address_space(1))) int gi32_t;  // global int
typedef __attribute__((address_space(3))) int li32_t;  // LDS int
#endif

// ---------------------------------------------------------------------------
// Pass 1: transpose one level's grid [C, H*W] -> [H*W, C] (64B records) in ws.
// Each block handles 256 pixels x 16 channels through LDS.
// Global->LDS fill uses the CDNA5 async-to-LDS path when available.
// ---------------------------------------------------------------------------
__global__ __launch_bounds__(256) void mrg_transpose(
    const float* __restrict__ g, float* __restrict__ ws, int HW) {
  __shared__ float tile[C_CH * TP_PAD];
  const int tid  = threadIdx.x;
  const int base = blockIdx.x * 256;

#ifdef USE_ASYNC_LDS
#pragma unroll
  for (int c = 0; c < C_CH; ++c) {
    const float* gp = g + (size_t)c * (size_t)HW + (size_t)(base + tid);
    const float* lp = &tile[c * TP_PAD + tid];
    __builtin_amdgcn_global_load_async_to_lds_b32(
        (gi32_t*)(unsigned long long)(size_t)gp,
        (li32_t*)(unsigned int)(size_t)lp,
        0, 0);
  }
  __builtin_amdgcn_s_wait_asynccnt(0);
#else
#pragma unroll
  for (int c = 0; c < C_CH; ++c) {
    tile[c * TP_PAD + tid] = g[(size_t)c * (size_t)HW + (size_t)(base + tid)];
  }
#endif
  __syncthreads();

  float* dst = ws + (size_t)base * C_CH;  // fully coalesced linear writes
#pragma unroll
  for (int i = 0; i < C_CH; ++i) {
    int j   = i * 256 + tid;   // 0..4095 over this block's 256*16 outputs
    int pix = j >> 4;
    int cc  = j & 15;
    dst[j] = tile[cc * TP_PAD + pix];
  }
}

// ---------------------------------------------------------------------------
// Pass 2: bilinear sample. 4 lanes per point; each lane owns 4 channels and
// issues 128-bit gathers (4 lanes together cover each 64B corner record).
// Output (320 MB stream) written with non-temporal stores to preserve the
// L2-resident grids (89.4 MB < 192 MB L2).
// ---------------------------------------------------------------------------
__global__ __launch_bounds__(256) void mrg_sample(
    const float* __restrict__ xy, const float* __restrict__ ws,
    float* __restrict__ out, int N) {
  const int t = blockIdx.x * 256 + threadIdx.x;
  const int p = t >> 2;
  if (p >= N) return;
  const int sub = t & 3;

  const v2f pt = __builtin_nontemporal_load((const v2f*)xy + p);

  const int    res[NLEV]     = {64, 128, 256, 512, 1024};
  const size_t baseRec[NLEV] = {0, 4096, 20480, 86016, 348160};

  float* orow = out + (size_t)p * (NLEV * C_CH);

#pragma unroll
  for (int l = 0; l < NLEV; ++l) {
    const int   W  = res[l];
    const float wm = (float)(W - 1);
    // match reference: clip((x+1)*0.5*(W-1), 0, W-1), align_corners + border
    float fx = fminf(fmaxf(((pt.x + 1.0f) * 0.5f) * wm, 0.0f), wm);
    float fy = fminf(fmaxf(((pt.y + 1.0f) * 0.5f) * wm, 0.0f), wm);
    int x0 = (int)fx;                 // fx >= 0 -> trunc == floor
    int y0 = (int)fy;
    float wx = fx - (float)x0;
    float wy = fy - (float)y0;
    int x1 = min(x0 + 1, W - 1);
    int y1 = min(y0 + 1, W - 1);

    const v4f* rec = (const v4f*)(ws + baseRec[l] * C_CH);
    const size_t r00 = ((size_t)(y0 * W + x0) << 2) + sub;
    const size_t r01 = ((size_t)(y0 * W + x1) << 2) + sub;
    const size_t r10 = ((size_t)(y1 * W + x0) << 2) + sub;
    const size_t r11 = ((size_t)(y1 * W + x1) << 2) + sub;
    v4f v00 = rec[r00];
    v4f v01 = rec[r01];
    v4f v10 = rec[r10];
    v4f v11 = rec[r11];

    // same association order as the reference expression
    v4f r = (v00 * (1.0f - wx) + v01 * wx) * (1.0f - wy) +
            (v10 * (1.0f - wx) + v11 * wx) * wy;

    __builtin_nontemporal_store(r, (v4f*)(orow + l * C_CH) + sub);
  }
}

// ---------------------------------------------------------------------------
// Fallback (only if ws is too small for the transposed grids): gather directly
// from [C, H, W] with scalar loads. Deterministic host-side selection.
// ---------------------------------------------------------------------------
__global__ __launch_bounds__(256) void mrg_sample_direct(
    const float* __restrict__ xy,
    const float* __restrict__ g0, const float* __restrict__ g1,
    const float* __restrict__ g2, const float* __restrict__ g3,
    const float* __restrict__ g4, float* __restrict__ out, int N) {
  const int t = blockIdx.x * 256 + threadIdx.x;
  const int p = t >> 2;
  if (p >= N) return;
  const int sub = t & 3;

  const v2f pt = __builtin_nontemporal_load((const v2f*)xy + p);
  const float* grids[NLEV] = {g0, g1, g2, g3, g4};
  const int res[NLEV] = {64, 128, 256, 512, 1024};
  float* orow = out + (size_t)p * (NLEV * C_CH);

#pragma unroll
  for (int l = 0; l < NLEV; ++l) {
    const int   W  = res[l];
    const int   HW = W * W;
    const float wm = (float)(W - 1);
    float fx = fminf(fmaxf(((pt.x + 1.0f) * 0.5f) * wm, 0.0f), wm);
    float fy = fminf(fmaxf(((pt.y + 1.0f) * 0.5f) * wm, 0.0f), wm);
    int x0 = (int)fx;
    int y0 = (int)fy;
    float wx = fx - (float)x0;
    float wy = fy - (float)y0;
    int x1 = min(x0 + 1, W - 1);
    int y1 = min(y0 + 1, W - 1);

    const float* g = grids[l];
    const size_t i00 = (size_t)(y0 * W + x0);
    const size_t i01 = (size_t)(y0 * W + x1);
    const size_t i10 = (size_t)(y1 * W + x0);
    const size_t i11 = (size_t)(y1 * W + x1);

    v4f r;
#pragma unroll
    for (int k = 0; k < 4; ++k) {
      const int c = sub * 4 + k;
      const float* gc = g + (size_t)c * (size_t)HW;
      float v00 = gc[i00], v01 = gc[i01], v10 = gc[i10], v11 = gc[i11];
      r[k] = (v00 * (1.0f - wx) + v01 * wx) * (1.0f - wy) +
             (v10 * (1.0f - wx) + v11 * wx) * wy;
    }
    __builtin_nontemporal_store(r, (v4f*)(orow + l * C_CH) + sub);
  }
}

extern "C" void kernel_launch(void* const* d_in, const int* in_sizes, int n_in,
                              void* d_out, int out_size, void* d_ws, size_t ws_size,
                              hipStream_t stream) {
  const float* xy = (const float*)d_in[0];
  const float* g[NLEV] = {(const float*)d_in[1], (const float*)d_in[2],
                          (const float*)d_in[3], (const float*)d_in[4],
                          (const float*)d_in[5]};
  float* out = (float*)d_out;
  const int N = in_sizes[0] / 2;

  const int    HW[NLEV]      = {4096, 16384, 65536, 262144, 1048576};
  const size_t baseRec[NLEV] = {0, 4096, 20480, 86016, 348160};
  const size_t needed = (size_t)(4096 + 16384 + 65536 + 262144 + 1048576) *
                        (size_t)C_CH * sizeof(float);  // 89,391,104 B

  const long long threads = 4LL * (long long)N;
  const int blocks = (int)((threads + 255) / 256);

  if (ws_size >= needed) {
    float* ws = (float*)d_ws;
    for (int l = 0; l < NLEV; ++l) {
      mrg_transpose<<<HW[l] / 256, 256, 0, stream>>>(g[l], ws + baseRec[l] * C_CH,
                                                     HW[l]);
    }
    mrg_sample<<<blocks, 256, 0, stream>>>(xy, ws, out, N);
  } else {
    mrg_sample_direct<<<blocks, 256, 0, stream>>>(xy, g[0], g[1], g[2], g[3],
                                                  g[4], out, N);
  }
}